// RNN_76527727280241
// MI455X (gfx1250) — compile-verified
//
#include <hip/hip_runtime.h>
#include <math.h>

// ---------------- problem constants ----------------
#define EMB 300
#define HID 10
#define BATCH 256
#define SEQ 512
#define M_TOTAL (BATCH * SEQ)   // 131072 rows of x viewed as [M, EMB]
#define GI_STRIDE 32            // 3H=30 padded to 32 for aligned indexing
#define MT 4                    // M-tiles (16 rows each) per wave in phase 1

typedef float v2f __attribute__((ext_vector_type(2)));
typedef float v4f __attribute__((ext_vector_type(4)));
typedef float v8f __attribute__((ext_vector_type(8)));

__device__ __forceinline__ float sigmoid_fast(float x) {
    return 1.0f / (1.0f + __expf(-x));
}
__device__ __forceinline__ float tanh_fast(float x) {
    float ax = fabsf(x);
    float e  = __expf(-2.0f * ax);
    float t  = (1.0f - e) / (1.0f + e);
    return copysignf(t, x);
}

// ============================================================
// Phase 1: gi = x @ w_ih_f^T + b_ih_f    ([M,300]x[300,30] fp32 WMMA)
// One wave = MT(=4) 16-row M tiles x both 16-wide N tiles (N padded 30->32).
// B loads amortized over 4 A tiles: 6 b64 loads -> 8 WMMAs per k-block.
// #pragma unroll 1 on the k-loop: prevents modulo-variable-expansion from
// duplicating the 8 accumulator tiles (round-2 disasm showed ~32 v_mov_b64
// per iteration shuttling accumulators between two register sets).
// ============================================================
__global__ __launch_bounds__(256) void gi_wmma_kernel(
    const float* __restrict__ x, const float* __restrict__ w_ih,
    const float* __restrict__ b_ih, float* __restrict__ gi)
{
    const int lane  = threadIdx.x & 31;
    const int wave  = threadIdx.x >> 5;
    const int mbase = (blockIdx.x * 8 + wave) * (16 * MT);
    const bool hi   = lane >= 16;
    const int l15   = lane & 15;
    const int n0    = l15;                                  // N-tile 0 column
    const int n1    = 16 + l15;                             // N-tile 1 column
    const int n1c   = (n1 < 3 * HID) ? n1 : (3 * HID - 1);  // safe address
    const int koff  = hi ? 2 : 0;   // A/B layout: lanes 16-31 hold K=k0+2,k0+3

    const float* ap[MT];
    #pragma unroll
    for (int m = 0; m < MT; ++m)
        ap[m] = x + (size_t)(mbase + 16 * m + l15) * EMB + koff;
    const float* b0p = w_ih + (size_t)n0  * EMB + koff;
    const float* b1p = w_ih + (size_t)n1c * EMB + koff;

    v8f c0[MT], c1[MT];
    #pragma unroll
    for (int m = 0; m < MT; ++m) { c0[m] = (v8f){}; c1[m] = (v8f){}; }

    #pragma unroll 1
    for (int k0 = 0; k0 < EMB; k0 += 4) {
        v2f b0 = *(const v2f*)(b0p + k0);
        v2f b1 = *(const v2f*)(b1p + k0);
        v2f a[MT];
        #pragma unroll
        for (int m = 0; m < MT; ++m) a[m] = *(const v2f*)(ap[m] + k0);
        #pragma unroll
        for (int m = 0; m < MT; ++m) {
            c0[m] = __builtin_amdgcn_wmma_f32_16x16x4_f32(
                        false, a[m], false, b0, (short)0, c0[m], false, false);
            c1[m] = __builtin_amdgcn_wmma_f32_16x16x4_f32(
                        false, a[m], false, b1, (short)0, c1[m], false, false);
        }
    }

    // epilogue: each lane's accumulators share one column n -> fuse bias
    float bias0 = b_ih[n0];
    float bias1 = (n1 < 3 * HID) ? b_ih[n1] : 0.0f;
    #pragma unroll
    for (int m = 0; m < MT; ++m) {
        const int rbase = mbase + 16 * m + (hi ? 8 : 0);  // C: VGPR r -> M=r / 8+r
        #pragma unroll
        for (int r = 0; r < 8; ++r) {
            size_t row = (size_t)(rbase + r);
            gi[row * GI_STRIDE + n0] = c0[m][r] + bias0;
            gi[row * GI_STRIDE + n1] = c1[m][r] + bias1;  // pad cols: never read
        }
    }
}

// ============================================================
// Phase 2: forward GRU recurrence. One wave per batch element.
// Latency-bound sequential chain -> minimize per-step dependence depth:
//  - pairwise-tree dot (depth ~5 vs 10-deep FMA chain)
//  - next step's gi loads prefetched before current gate math
// ============================================================
__global__ __launch_bounds__(256) void gru_fwd_recurrence(
    const float* __restrict__ gi, const float* __restrict__ w_hh,
    const float* __restrict__ b_hh, float* __restrict__ hf_out)
{
    const int lane = threadIdx.x & 31;
    const int wave = threadIdx.x >> 5;
    const int b = blockIdx.x * 8 + wave;         // batch element
    const int j = (lane < HID) ? lane : 0;       // clamped hidden index

    float wr[HID], wz[HID], wn[HID];
    #pragma unroll
    for (int k = 0; k < HID; ++k) {
        wr[k] = w_hh[(0 * HID + j) * HID + k];
        wz[k] = w_hh[(1 * HID + j) * HID + k];
        wn[k] = w_hh[(2 * HID + j) * HID + k];
    }
    const float br = b_hh[0 * HID + j];
    const float bz = b_hh[1 * HID + j];
    const float bn = b_hh[2 * HID + j];

    float h[HID];
    #pragma unroll
    for (int k = 0; k < HID; ++k) h[k] = 0.0f;
    float hown = 0.0f;

    const float* gib = gi + (size_t)b * SEQ * GI_STRIDE;
    float ir = gib[j], iz = gib[HID + j], in_ = gib[2 * HID + j];

    for (int t = 0; t < SEQ; ++t) {
        // prefetch next step's input gates (independent of h)
        int tn = (t + 1 < SEQ) ? (t + 1) : (SEQ - 1);
        const float* gnx = gib + (size_t)tn * GI_STRIDE;
        float irn = gnx[j], izn = gnx[HID + j], inn = gnx[2 * HID + j];

        // three dot-10s as mult + pairwise tree (short critical path)
        float pr[HID], pz[HID], pn[HID];
        #pragma unroll
        for (int k = 0; k < HID; ++k) {
            pr[k] = wr[k] * h[k];
            pz[k] = wz[k] * h[k];
            pn[k] = wn[k] * h[k];
        }
        float gr = (((pr[0]+pr[1]) + (pr[2]+pr[3])) + ((pr[4]+pr[5]) + (pr[6]+pr[7])))
                   + ((pr[8]+pr[9]) + br);
        float gz = (((pz[0]+pz[1]) + (pz[2]+pz[3])) + ((pz[4]+pz[5]) + (pz[6]+pz[7])))
                   + ((pz[8]+pz[9]) + bz);
        float gn = (((pn[0]+pn[1]) + (pn[2]+pn[3])) + ((pn[4]+pn[5]) + (pn[6]+pn[7])))
                   + ((pn[8]+pn[9]) + bn);

        float r  = sigmoid_fast(ir + gr);
        float z  = sigmoid_fast(iz + gz);
        float n  = tanh_fast(fmaf(r, gn, in_));
        float hj = (1.0f - z) * n + z * hown;

        #pragma unroll
        for (int k = 0; k < HID; ++k) h[k] = __shfl(hj, k, 32);
        hown = hj;
        ir = irn; iz = izn; in_ = inn;
    }
    if (lane < HID) hf_out[b * HID + lane] = hown;
}

// ============================================================
// Phase 3: backward-dir single step (h0 = 0 -> gh = b_hh_b) + linear head.
// One wave per batch element; lanes 0-29 each compute one gate dot-300
// with float4 loads (rows are 16B-aligned: 300 % 4 == 0).
// ============================================================
__global__ __launch_bounds__(256) void gru_bwd_and_head(
    const float* __restrict__ x, const float* __restrict__ w_ih_b,
    const float* __restrict__ b_ih_b, const float* __restrict__ b_hh_b,
    const float* __restrict__ w_lin, const float* __restrict__ b_lin,
    const float* __restrict__ hf_final, float* __restrict__ out)
{
    const int lane = threadIdx.x & 31;
    const int wave = threadIdx.x >> 5;
    const int b = blockIdx.x * 8 + wave;
    const int g = (lane < 3 * HID) ? lane : 0;

    const float* xr = x + ((size_t)b * SEQ + (SEQ - 1)) * EMB;  // x[b, S-1, :]
    const float* wg = w_ih_b + (size_t)g * EMB;
    float acc = b_ih_b[g];
    for (int k = 0; k < EMB; k += 4) {
        v4f wv = *(const v4f*)(wg + k);
        v4f xv = *(const v4f*)(xr + k);
        acc = fmaf(wv.x, xv.x, acc);
        acc = fmaf(wv.y, xv.y, acc);
        acc = fmaf(wv.z, xv.z, acc);
        acc = fmaf(wv.w, xv.w, acc);
    }

    const int j = (lane < HID) ? lane : 0;
    float i_r = __shfl(acc, j, 32);
    float i_z = __shfl(acc, HID + j, 32);
    float i_n = __shfl(acc, 2 * HID + j, 32);

    float r  = sigmoid_fast(i_r + b_hh_b[j]);
    float z  = sigmoid_fast(i_z + b_hh_b[HID + j]);
    float n  = tanh_fast(fmaf(r, b_hh_b[2 * HID + j], i_n));
    float hb = (1.0f - z) * n;                   // h0 = 0 => z*h0 vanishes

    float partial = 0.0f;
    if (lane < HID)
        partial = fmaf(w_lin[lane], hf_final[b * HID + lane],
                       w_lin[HID + lane] * hb);
    #pragma unroll
    for (int off = 16; off >= 1; off >>= 1)
        partial += __shfl_xor(partial, off, 32);
    if (lane == 0) out[b] = partial + b_lin[0];
}

// ============================================================
extern "C" void kernel_launch(void* const* d_in, const int* in_sizes, int n_in,
                              void* d_out, int out_size, void* d_ws, size_t ws_size,
                              hipStream_t stream)
{
    (void)in_sizes; (void)n_in; (void)out_size; (void)ws_size;
    const float* x      = (const float*)d_in[0];
    const float* w_ih_f = (const float*)d_in[1];
    const float* w_hh_f = (const float*)d_in[2];
    const float* b_ih_f = (const float*)d_in[3];
    const float* b_hh_f = (const float*)d_in[4];
    const float* w_ih_b = (const float*)d_in[5];
    // d_in[6] = w_hh_b: provably unused (backward state uses h0 = 0, one step)
    const float* b_ih_b = (const float*)d_in[7];
    const float* b_hh_b = (const float*)d_in[8];
    const float* w_lin  = (const float*)d_in[9];
    const float* b_lin  = (const float*)d_in[10];
    float* out = (float*)d_out;

    float* gi       = (float*)d_ws;                         // M_TOTAL*32 f32 = 16.8 MB
    float* hf_final = gi + (size_t)M_TOTAL * GI_STRIDE;     // 256*10 f32

    gi_wmma_kernel<<<dim3(M_TOTAL / (16 * MT * 8)), dim3(256), 0, stream>>>(
        x, w_ih_f, b_ih_f, gi);
    gru_fwd_recurrence<<<dim3(BATCH / 8), dim3(256), 0, stream>>>(
        gi, w_hh_f, b_hh_f, hf_final);
    gru_bwd_and_head<<<dim3(BATCH / 8), dim3(256), 0, stream>>>(
        x, w_ih_b, b_ih_b, b_hh_b, w_lin, b_lin, hf_final, out);
}